// LIConv2dWeights_27075473834357
// MI455X (gfx1250) — compile-verified
//
#include <hip/hip_runtime.h>
#include <hip/hip_bf16.h>

typedef __attribute__((ext_vector_type(2))) float v2f;
typedef __attribute__((ext_vector_type(8))) float v8f;
typedef __attribute__((ext_vector_type(4))) unsigned int v4u;
typedef __attribute__((ext_vector_type(4))) int v4i;
typedef __attribute__((ext_vector_type(8))) int v8i;

constexpr int Fd = 576;     // in_features == inducing batch
constexpr int Od = 256;     // out channels
constexpr int Sd = 64;      // samples
constexpr int NC = Sd * Od; // 16384 RHS rows

// ---------------------------------------------------------------------------
// 1) Wb[i,f] = exp(log_prec[i]) * B[i,f]
// ---------------------------------------------------------------------------
__global__ __launch_bounds__(256) void scale_kernel(const float* __restrict__ B,
                                                    const float* __restrict__ lp,
                                                    float* __restrict__ Wb) {
  int idx = blockIdx.x * blockDim.x + threadIdx.x;
  if (idx < Fd * Fd) {
    int i = idx / Fd;
    Wb[idx] = __expf(lp[i]) * B[idx];
  }
}

// ---------------------------------------------------------------------------
// 2) Generic strided fp32 WMMA GEMM: C[m,n] = sum_k A[m,k]*B[k,n] (+diag_add on m==n)
//    A[m,k] at A[k*a_ks + m*a_ms]; B[k,n] at B[k*b_ks + n*b_ns]
//    One wave per 16x16 tile; V_WMMA_F32_16X16X4_F32, K%4==0, M,N%16==0.
// ---------------------------------------------------------------------------
__global__ __launch_bounds__(128) void wmma_gemm_generic(
    const float* __restrict__ A, int a_ks, int a_ms,
    const float* __restrict__ B, int b_ks, int b_ns,
    float* __restrict__ C, int ldc, int M, int N, int K, float diag_add) {
  int lane = threadIdx.x & 31;
  int w    = threadIdx.x >> 5;
  int m0   = blockIdx.x * 16;
  int n0   = (blockIdx.y * 4 + w) * 16;
  if (m0 >= M || n0 >= N) return;  // wave-uniform

  int l16  = lane & 15;
  int koff = (lane >= 16) ? 2 : 0;
  int am   = m0 + l16;
  int bn   = n0 + l16;

  v8f acc = {};
  for (int k = 0; k < K; k += 4) {
    int k0 = k + koff;
    v2f av, bv;
    av.x = A[k0 * a_ks + am * a_ms];
    av.y = A[(k0 + 1) * a_ks + am * a_ms];
    bv.x = B[k0 * b_ks + bn * b_ns];
    bv.y = B[(k0 + 1) * b_ks + bn * b_ns];
    acc = __builtin_amdgcn_wmma_f32_16x16x4_f32(false, av, false, bv,
                                                (short)0, acc, false, false);
  }
  int hi8 = (lane >= 16) ? 8 : 0;
  for (int v = 0; v < 8; ++v) {
    int mm = m0 + v + hi8;
    int nn = n0 + l16;
    float val = acc[v];
    if (mm == nn) val += diag_add;
    C[mm * ldc + nn] = val;
  }
}

// ---------------------------------------------------------------------------
// 3) In-place Cholesky of P (uses/produces lower triangle), single workgroup.
//    Also writes logdet = 2*sum(log diag(L)).
// ---------------------------------------------------------------------------
__global__ __launch_bounds__(1024) void cholesky_kernel(float* __restrict__ P,
                                                        float* __restrict__ logdet_out) {
  __shared__ float col[Fd];
  __shared__ float sd;
  int tid = threadIdx.x;
  int nt  = blockDim.x;
  float logacc = 0.f;
  for (int k = 0; k < Fd; ++k) {
    if (tid == 0) {
      float dv = sqrtf(P[k * Fd + k]);
      P[k * Fd + k] = dv;
      sd = dv;
      logacc += 2.f * __logf(dv);
    }
    __syncthreads();
    float dv = sd;
    for (int i = k + 1 + tid; i < Fd; i += nt) {
      float v = P[i * Fd + k] / dv;
      P[i * Fd + k] = v;
      col[i] = v;
    }
    __syncthreads();
    int R = Fd - k - 1;
    for (int idx = tid; idx < R * R; idx += nt) {
      int i = k + 1 + idx / R;
      int j = k + 1 + idx % R;
      if (j <= i) P[i * Fd + j] -= col[i] * col[j];
    }
    __syncthreads();
  }
  if (tid == 0) *logdet_out = logacc;
}

// ---------------------------------------------------------------------------
// 4) UinvT[g,f] = (L^-T)[f,g], one wave per column j of Uinv (back-substitution
//    on U = L^T). Column buffered in LDS; zero fill below diagonal.
// ---------------------------------------------------------------------------
__global__ __launch_bounds__(256) void uinv_kernel(const float* __restrict__ L,
                                                   float* __restrict__ UT) {
  __shared__ float xb[8][Fd];
  int w = threadIdx.x >> 5, lane = threadIdx.x & 31;
  int j = blockIdx.x * 8 + w;
  float* x = xb[w];
  for (int i = lane; i < Fd; i += 32) x[i] = 0.f;
  if (lane == 0) x[j] = 1.0f / L[j * Fd + j];
  for (int i = j - 1; i >= 0; --i) {
    float s = 0.f;
    for (int m = i + 1 + lane; m <= j; m += 32)
      s += L[m * Fd + i] * x[m];  // U[i,m] = L[m,i]
    for (int off = 16; off > 0; off >>= 1) s += __shfl_down(s, off, 32);
    if (lane == 0) x[i] = -s / L[i * Fd + i];
  }
  for (int i = lane; i < Fd; i += 32) UT[j * Fd + i] = x[i];
}

// ---------------------------------------------------------------------------
// 6) Main GEMM: Out[r,f] = sum_g Z[r,g]*UinvT[g,f] + mean[f, r%256]
//    Block: 256 threads / 8 waves, tile 128(M)x64(N), WMMA fp32.
//    A tile: float4 global->LDS staging (padded stride 20).
//    B tile: TENSOR_LOAD_TO_LDS (TDM DMA, issued by wave 0, s_wait_tensorcnt).
//    Also emits per-block partials of sum(sample^2) and (by==0) sum(Z^2).
// ---------------------------------------------------------------------------
constexpr int BM = 128, BN = 64, BK = 16, LDP = BK + 4;  // stride 20: 16B-aligned, bank-safe

__global__ __launch_bounds__(256) void sample_gemm(
    const float* __restrict__ Zf, const float* __restrict__ UT,
    const float* __restrict__ mean, float* __restrict__ out,
    float* __restrict__ ss_part, float* __restrict__ zz_part) {
  __shared__ float As[BM][LDP];
  __shared__ float Bs[BK * BN];  // TDM writes tile row-major contiguous: Bs[k*BN + n]
  __shared__ float red[16];
  int tid = threadIdx.x;
  int lane = tid & 31, w = tid >> 5;
  int bx = blockIdx.x, by = blockIdx.y;
  int wm = w & 3, wn = w >> 2;
  int r0 = bx * BM;
  int n0 = by * BN;

  v8f acc[2][2] = {{{}, {}}, {{}, {}}};
  float zloc = 0.f;

  int tm = tid >> 1, th = tid & 1;  // A-tile loader: row tm, half th

  // Invariant parts of the Tensor DMA descriptor (2-D tensor, fp32).
  unsigned ldsBsOff = (unsigned)(unsigned long long)&Bs[0];
  v8i g1;
  g1.s0 = 0x00020000;                               // data_size = 4 bytes
  g1.s1 = (Fd & 0xffff) << 16;                      // tensor_dim0[15:0]
  g1.s2 = ((Fd >> 16) & 0xffff) | ((Fd & 0xffff) << 16);  // dim0[31:16] | dim1[15:0]
  g1.s3 = ((Fd >> 16) & 0xffff) | (BN << 16);       // dim1[31:16] | tile_dim0
  g1.s4 = BK;                                       // tile_dim1 | tile_dim2=0
  g1.s5 = Fd;                                       // tensor_dim0_stride[31:0]
  g1.s6 = 0;
  g1.s7 = 0;
  v4i gz = {0, 0, 0, 0};
  v8i gz8 = {0, 0, 0, 0, 0, 0, 0, 0};

  for (int kk = 0; kk < Fd; kk += BK) {
    // ---- B tile via TDM: UT[kk..kk+15, n0..n0+63] -> Bs (wave 0 only) ----
    if (w == 0) {
      unsigned long long gaddr =
          (unsigned long long)(const void*)(UT + (long)kk * Fd + n0);
      v4u g0;
      g0.x = 1u;  // count=1, user mode
      g0.y = ldsBsOff;
      g0.z = (unsigned)gaddr;
      g0.w = (unsigned)((gaddr >> 32) & 0x01FFFFFFull) | (2u << 30);  // type=2
#if __clang_major__ >= 23
      __builtin_amdgcn_tensor_load_to_lds(g0, g1, gz, gz, gz8, 0);
#else
      __builtin_amdgcn_tensor_load_to_lds(g0, g1, gz, gz, 0);
#endif
    }

    // ---- A tile: Z rows r0..r0+127, cols kk..kk+15 ----
    const float4* src = reinterpret_cast<const float4*>(
        Zf + (long)(r0 + tm) * Fd + kk + th * 8);
    float4 p0 = src[0];
    float4 p1 = src[1];
    __builtin_prefetch(Zf + (long)(r0 + tm) * Fd + kk + BK + th * 8, 0, 3);
    *reinterpret_cast<float4*>(&As[tm][th * 8])     = p0;
    *reinterpret_cast<float4*>(&As[tm][th * 8 + 4]) = p1;
    if (by == 0) {
      zloc += p0.x * p0.x + p0.y * p0.y + p0.z * p0.z + p0.w * p0.w;
      zloc += p1.x * p1.x + p1.y * p1.y + p1.z * p1.z + p1.w * p1.w;
    }

    if (w == 0) __builtin_amdgcn_s_wait_tensorcnt(0);
    __syncthreads();

    int l16  = lane & 15;
    int koff = (lane >= 16) ? 2 : 0;
#pragma unroll
    for (int ks = 0; ks < BK; ks += 4) {
      int kA = ks + koff;
      v2f a0 = *reinterpret_cast<const v2f*>(&As[wm * 32 + l16][kA]);
      v2f a1 = *reinterpret_cast<const v2f*>(&As[wm * 32 + 16 + l16][kA]);
      v2f b0, b1;
      b0.x = Bs[kA * BN + wn * 32 + l16];
      b0.y = Bs[(kA + 1) * BN + wn * 32 + l16];
      b1.x = Bs[kA * BN + wn * 32 + 16 + l16];
      b1.y = Bs[(kA + 1) * BN + wn * 32 + 16 + l16];
      acc[0][0] = __builtin_amdgcn_wmma_f32_16x16x4_f32(false, a0, false, b0, (short)0, acc[0][0], false, false);
      acc[0][1] = __builtin_amdgcn_wmma_f32_16x16x4_f32(false, a0, false, b1, (short)0, acc[0][1], false, false);
      acc[1][0] = __builtin_amdgcn_wmma_f32_16x16x4_f32(false, a1, false, b0, (short)0, acc[1][0], false, false);
      acc[1][1] = __builtin_amdgcn_wmma_f32_16x16x4_f32(false, a1, false, b1, (short)0, acc[1][1], false, false);
    }
    __syncthreads();
  }

  // Epilogue: add mean, write sample, accumulate sum(sample^2)
  float ssloc = 0.f;
  int l16 = lane & 15;
  int hi8 = (lane >= 16) ? 8 : 0;
#pragma unroll
  for (int mi = 0; mi < 2; ++mi)
#pragma unroll
    for (int ni = 0; ni < 2; ++ni)
#pragma unroll
      for (int v = 0; v < 8; ++v) {
        int rr = r0 + wm * 32 + mi * 16 + hi8 + v;
        int ff = n0 + wn * 32 + ni * 16 + l16;
        int oo = rr & (Od - 1);
        float val = acc[mi][ni][v] + mean[ff * Od + oo];
        out[(long)rr * Fd + ff] = val;
        ssloc += val * val;
      }
  for (int off = 16; off > 0; off >>= 1) ssloc += __shfl_down(ssloc, off, 32);
  if (lane == 0) red[w] = ssloc;
  __syncthreads();
  if (tid == 0) {
    float t = 0.f;
    for (int i = 0; i < 8; ++i) t += red[i];
    ss_part[bx * 9 + by] = t;
  }
  if (by == 0) {
    for (int off = 16; off > 0; off >>= 1) zloc += __shfl_down(zloc, off, 32);
    if (lane == 0) red[8 + w] = zloc;
    __syncthreads();
    if (tid == 0) {
      float t = 0.f;
      for (int i = 0; i < 8; ++i) t += red[8 + i];
      zz_part[bx] = t;
    }
  }
}

// ---------------------------------------------------------------------------
// 7) logpq[s] from deterministic partials
// ---------------------------------------------------------------------------
__global__ __launch_bounds__(64) void finalize_kernel(
    const float* __restrict__ ss_part, const float* __restrict__ zz_part,
    const float* __restrict__ logdet_p, float* __restrict__ out_logpq) {
  int s = threadIdx.x;
  if (s >= Sd) return;
  float ss = 0.f;
  for (int b = 0; b < 2; ++b)
    for (int n = 0; n < 9; ++n) ss += ss_part[(2 * s + b) * 9 + n];
  float zz = zz_part[2 * s] + zz_part[2 * s + 1];
  float logdet = *logdet_p;
  float logP = -0.5f * (float)Fd * ss + 0.5f * (float)Od * (float)Fd * __logf((float)Fd);
  float logQ = -0.5f * zz + 0.5f * (float)Od * logdet;
  out_logpq[s] = logP - logQ;
}

// ---------------------------------------------------------------------------
// Host orchestration
// ---------------------------------------------------------------------------
extern "C" void kernel_launch(void* const* d_in, const int* in_sizes, int n_in,
                              void* d_out, int out_size, void* d_ws, size_t ws_size,
                              hipStream_t stream) {
  const float* Xi_param = (const float*)d_in[1];  // [1, F, F]
  const float* u        = (const float*)d_in[2];  // [O, F, 1]
  const float* lp       = (const float*)d_in[3];  // [1, 1, F]
  const float* Z        = (const float*)d_in[4];  // [S, O, F, 1]
  float* out = (float*)d_out;
  float* ws  = (float*)d_ws;

  const size_t o_Wb     = 0;
  const size_t o_L      = o_Wb + (size_t)Fd * Fd;
  const size_t o_UT     = o_L + (size_t)Fd * Fd;
  const size_t o_XiLY   = o_UT + (size_t)Fd * Fd;
  const size_t o_tmp    = o_XiLY + (size_t)Fd * Od;
  const size_t o_mean   = o_tmp + (size_t)Fd * Od;
  const size_t o_logdet = o_mean + (size_t)Fd * Od;
  const size_t o_ss     = o_logdet + 1;
  const size_t o_zz     = o_ss + 128 * 9;

  // 1) weighted matrix
  scale_kernel<<<(Fd * Fd + 255) / 256, 256, 0, stream>>>(Xi_param, lp, ws + o_Wb);

  // 2) prec = Wb^T B + F*I   (A[m,k]=Wb[k*F+m]; B[k,n]=Bmat[k*F+n])
  wmma_gemm_generic<<<dim3(Fd / 16, 9), 128, 0, stream>>>(
      ws + o_Wb, Fd, 1, Xi_param, Fd, 1, ws + o_L, Fd, Fd, Fd, Fd, (float)Fd);
  //    XiLY = Wb^T u^T       (B[k,n]=u[n*F+k])
  wmma_gemm_generic<<<dim3(Fd / 16, 4), 128, 0, stream>>>(
      ws + o_Wb, Fd, 1, u, 1, Fd, ws + o_XiLY, Od, Fd, Od, Fd, 0.f);

  // 3) Cholesky (in place, lower) + logdet
  cholesky_kernel<<<1, 1024, 0, stream>>>(ws + o_L, ws + o_logdet);

  // 4) UinvT[g,f] = (L^-T)[f,g]
  uinv_kernel<<<Fd / 8, 256, 0, stream>>>(ws + o_L, ws + o_UT);

  // 5) tmp = L^-1 XiLY       (A[m,k]=UT[m*F+k])
  wmma_gemm_generic<<<dim3(Fd / 16, 4), 128, 0, stream>>>(
      ws + o_UT, 1, Fd, ws + o_XiLY, Od, 1, ws + o_tmp, Od, Fd, Od, Fd, 0.f);
  //    mean = Uinv tmp       (A[m,k]=UT[k*F+m])
  wmma_gemm_generic<<<dim3(Fd / 16, 4), 128, 0, stream>>>(
      ws + o_UT, Fd, 1, ws + o_tmp, Od, 1, ws + o_mean, Od, Fd, Od, Fd, 0.f);

  // 6) sample = Z @ Uinv^T + mean ; partial reductions
  sample_gemm<<<dim3(NC / BM, Fd / BN), 256, 0, stream>>>(
      Z, ws + o_UT, ws + o_mean, out, ws + o_ss, ws + o_zz);

  // 7) logpq
  finalize_kernel<<<1, 64, 0, stream>>>(ws + o_ss, ws + o_zz, ws + o_logdet,
                                        out + (size_t)Sd * Od * Fd);
}